// DREMVCL_46213848104976
// MI455X (gfx1250) — compile-verified
//
#include <hip/hip_runtime.h>
#include <math.h>

// ---------------- problem constants ----------------
#define DRUG_NUM 8000
#define DIS_NUM  8000
#define DD       64
#define NNZ_     1000000
#define BB       4096
#define INV_T    20.0f   // 1/0.05

typedef float v2f __attribute__((ext_vector_type(2)));
typedef float v8f __attribute__((ext_vector_type(8)));
typedef int   v4i __attribute__((ext_vector_type(4)));

#define AS1 __attribute__((address_space(1)))
#define AS3 __attribute__((address_space(3)))

#if __has_builtin(__builtin_amdgcn_global_load_async_to_lds_b128)
#define HAS_ASYNC_LDS 1
#else
#define HAS_ASYNC_LDS 0
#endif

// CDNA5 fp32 WMMA: D(16x16,f32) = A(16x4,f32) x B(4x16,f32) + C
__device__ __forceinline__ v8f wmma4(v2f a, v2f b, v8f c) {
  return __builtin_amdgcn_wmma_f32_16x16x4_f32(false, a, false, b, (short)0, c,
                                               false, false);
}

// async global->LDS 16B copy (gfx1250 GLOBAL_LOAD_ASYNC_TO_LDS_B128)
// builtin signature: (v4i addrspace(1)* src, v4i addrspace(3)* dst, Ii off, Ii cpol)
__device__ __forceinline__ void async_ld16(const float* g, float* l) {
#if HAS_ASYNC_LDS
  __builtin_amdgcn_global_load_async_to_lds_b128((AS1 v4i*)g, (AS3 v4i*)l, 0, 0);
#endif
}

template <int N>
__device__ __forceinline__ void wait_async() {
#if HAS_ASYNC_LDS
#if __has_builtin(__builtin_amdgcn_s_wait_asynccnt)
  __builtin_amdgcn_s_wait_asynccnt(N);
#else
  asm volatile("s_wait_asynccnt %0" ::"i"(N) : "memory");
#endif
#endif
}

__device__ __forceinline__ float wred32(float v) {
  v += __shfl_xor(v, 1);
  v += __shfl_xor(v, 2);
  v += __shfl_xor(v, 4);
  v += __shfl_xor(v, 8);
  v += __shfl_xor(v, 16);
  return v;
}

// ---------------- utility kernels ----------------
__global__ void k_zero(float* __restrict__ p, int n) {
  int i = blockIdx.x * blockDim.x + threadIdx.x;
  if (i < n) p[i] = 0.0f;
}

// Fused SpMM for A and A^T: one wave per edge, lane handles 2 dims.
// dstR[row] += v * srcR[col];  dstC[col] += v * srcC[row]
__global__ void k_spmm(const int* __restrict__ rows, const int* __restrict__ cols,
                       const float* __restrict__ vals,
                       const float* __restrict__ srcR, const float* __restrict__ srcC,
                       float* __restrict__ dstR, float* __restrict__ dstC, int nnz) {
  int gw   = (blockIdx.x * blockDim.x + threadIdx.x) >> 5;
  int lane = threadIdx.x & 31;
  if (gw >= nnz) return;
  int   r = rows[gw];
  int   c = cols[gw];
  float v = vals[gw];
  int   d = lane * 2;
  float2 a = *(const float2*)(srcR + (size_t)c * DD + d);
  float2 b = *(const float2*)(srcC + (size_t)r * DD + d);
  atomicAdd(dstR + (size_t)r * DD + d,     v * a.x);
  atomicAdd(dstR + (size_t)r * DD + d + 1, v * a.y);
  atomicAdd(dstC + (size_t)c * DD + d,     v * b.x);
  atomicAdd(dstC + (size_t)c * DD + d + 1, v * b.y);
}

// X = Ed0 + Zd0 ; SumEd = X ; Y = Er0 + Zr0 ; SumEr = Y
__global__ void k_addinit(const float* __restrict__ Er0, const float* __restrict__ Ed0,
                          const float* __restrict__ Zr0, const float* __restrict__ Zd0,
                          float* __restrict__ X, float* __restrict__ Y,
                          float* __restrict__ SumEr, float* __restrict__ SumEd, int n) {
  int i = blockIdx.x * blockDim.x + threadIdx.x;
  if (i < n) {
    float x = Ed0[i] + Zd0[i];
    float y = Er0[i] + Zr0[i];
    X[i] = x; SumEd[i] = x;
    Y[i] = y; SumEr[i] = y;
  }
}

// ---------------- dense GEMM over rec (WMMA + async double-buffered LDS) ----
// out[0:4096,0:64] = bias + (TRA ? rec^T : rec) @ V     (V is [8000 x 64])
// Block: 128 threads (4 waves); block tile 64 rows x 64 cols; K chunks of 64.
// Tiles are staged RAW (memory layout); the ^T case reads A column-wise.
#define LDSP 68
template <bool TRA>
__global__ __launch_bounds__(128)
void k_gemm(const float* __restrict__ rec, const float* __restrict__ V,
            const float* __restrict__ bias, float* __restrict__ out) {
  __shared__ __align__(16) float ldsA[2][64 * LDSP];
  __shared__ __align__(16) float ldsB[2][64 * LDSP];
  const int tid  = threadIdx.x;
  const int wv   = tid >> 5;
  const int lane = tid & 31;
  const int half = lane >> 4;
  const int l16  = lane & 15;
  const int m0   = blockIdx.x * 64;       // output row block (< 4096)
  const int c4   = (tid & 15) * 4;        // 0..60, float4 column
  const int r8   = tid >> 4;              // 0..7

  // stage one 64x64 rec tile + one 64x64 V chunk into LDS buffer `buf`
  auto stage = [&](int buf, int kb) {
#if HAS_ASYNC_LDS
    for (int rr = r8; rr < 64; rr += 8) {
      const float* g = TRA ? (rec + (size_t)(kb + rr) * DIS_NUM + m0 + c4)
                           : (rec + (size_t)(m0 + rr) * DIS_NUM + kb + c4);
      async_ld16(g, &ldsA[buf][rr * LDSP + c4]);
    }
    for (int rr = r8; rr < 64; rr += 8)
      async_ld16(V + (size_t)(kb + rr) * DD + c4, &ldsB[buf][rr * LDSP + c4]);
#else
    for (int rr = r8; rr < 64; rr += 8) {
      const float* g = TRA ? (rec + (size_t)(kb + rr) * DIS_NUM + m0 + c4)
                           : (rec + (size_t)(m0 + rr) * DIS_NUM + kb + c4);
      *(float4*)&ldsA[buf][rr * LDSP + c4] = *(const float4*)g;
    }
    for (int rr = r8; rr < 64; rr += 8)
      *(float4*)&ldsB[buf][rr * LDSP + c4] =
          *(const float4*)(V + (size_t)(kb + rr) * DD + c4);
#endif
  };

  v8f acc0 = {}, acc1 = {}, acc2 = {}, acc3 = {};

  stage(0, 0);
  for (int kb = 0; kb < DIS_NUM; kb += 64) {
    const int  cur  = (kb >> 6) & 1;
    const bool more = (kb + 64 < DIS_NUM);
    if (more) {
      stage(cur ^ 1, kb + 64);           // prefetch next tile into other buffer
      wait_async<16>();                   // in-order: <=16 left => current done
    } else {
      wait_async<0>();
    }
    __syncthreads();

    const float* A0 = &ldsA[cur][0];
    const float* B0 = &ldsB[cur][0];
#pragma unroll 4
    for (int k = 0; k < 64; k += 4) {
      const int kk = k + 2 * half;
      v2f a;
      if (TRA) {   // A[M=j][K=i] = raw[i][j]
        a.x = A0[kk * LDSP + wv * 16 + l16];
        a.y = A0[(kk + 1) * LDSP + wv * 16 + l16];
      } else {
        a = *(const v2f*)(A0 + (size_t)(wv * 16 + l16) * LDSP + kk);
      }
      v2f b0, b1, b2, b3;
      b0.x = B0[kk * LDSP +  0 + l16]; b0.y = B0[(kk + 1) * LDSP +  0 + l16];
      b1.x = B0[kk * LDSP + 16 + l16]; b1.y = B0[(kk + 1) * LDSP + 16 + l16];
      b2.x = B0[kk * LDSP + 32 + l16]; b2.y = B0[(kk + 1) * LDSP + 32 + l16];
      b3.x = B0[kk * LDSP + 48 + l16]; b3.y = B0[(kk + 1) * LDSP + 48 + l16];
      acc0 = wmma4(a, b0, acc0);
      acc1 = wmma4(a, b1, acc1);
      acc2 = wmma4(a, b2, acc2);
      acc3 = wmma4(a, b3, acc3);
    }
    __syncthreads();
  }

  // epilogue: C/D layout -> element (M = r + 8*half, N = l16) per VGPR r
  const int mrow = m0 + wv * 16 + 8 * half;
#pragma unroll
  for (int r = 0; r < 8; ++r) {
    const size_t rb = (size_t)(mrow + r) * DD;
    out[rb +  0 + l16] = acc0[r] + bias[rb +  0 + l16];
    out[rb + 16 + l16] = acc1[r] + bias[rb + 16 + l16];
    out[rb + 32 + l16] = acc2[r] + bias[rb + 32 + l16];
    out[rb + 48 + l16] = acc3[r] + bias[rb + 48 + l16];
  }
}

// ---------------- fused per-row kernel ----------------
__global__ void k_rowfuse(const float* __restrict__ SumEr, const float* __restrict__ SumGr,
                          const float* __restrict__ SumEd, const float* __restrict__ SumGd,
                          const float* __restrict__ labels,
                          const int* __restrict__ drugs, const int* __restrict__ dis,
                          float* __restrict__ N1r, float* __restrict__ N2r,
                          float* __restrict__ N1d, float* __restrict__ N2d,
                          float* __restrict__ posr, float* __restrict__ posd,
                          float* __restrict__ bce_acc, float* __restrict__ out_sig) {
  int gw   = (blockIdx.x * blockDim.x + threadIdx.x) >> 5;
  int lane = threadIdx.x & 31;
  if (gw >= BB) return;
  int rr = drugs[gw];
  int rd = dis[gw];
  int d  = lane * 2;
  float2 er = *(const float2*)(SumEr + (size_t)rr * DD + d);
  float2 gr = *(const float2*)(SumGr + (size_t)rr * DD + d);
  float2 ed = *(const float2*)(SumEd + (size_t)rd * DD + d);
  float2 gd = *(const float2*)(SumGd + (size_t)rd * DD + d);

  float sx = (er.x + gr.x) * (ed.x + gd.x) + (er.y + gr.y) * (ed.y + gd.y);
  float n1 = er.x * er.x + er.y * er.y;
  float n2 = gr.x * gr.x + gr.y * gr.y;
  float n3 = ed.x * ed.x + ed.y * ed.y;
  float n4 = gd.x * gd.x + gd.y * gd.y;
  float c1 = er.x * gr.x + er.y * gr.y;
  float c2 = ed.x * gd.x + ed.y * gd.y;
  sx = wred32(sx);
  n1 = wred32(n1); n2 = wred32(n2); n3 = wred32(n3); n4 = wred32(n4);
  c1 = wred32(c1); c2 = wred32(c2);
  float i1 = rsqrtf(n1), i2 = rsqrtf(n2), i3 = rsqrtf(n3), i4 = rsqrtf(n4);

  float2 o;
  o.x = er.x * i1; o.y = er.y * i1; *(float2*)(N1r + (size_t)gw * DD + d) = o;
  o.x = gr.x * i2; o.y = gr.y * i2; *(float2*)(N2r + (size_t)gw * DD + d) = o;
  o.x = ed.x * i3; o.y = ed.y * i3; *(float2*)(N1d + (size_t)gw * DD + d) = o;
  o.x = gd.x * i4; o.y = gd.y * i4; *(float2*)(N2d + (size_t)gw * DD + d) = o;

  if (lane == 0) {
    posr[gw] = c1 * i1 * i2 * INV_T;
    posd[gw] = c2 * i3 * i4 * INV_T;
    float s   = 0.25f * sx;                 // (0.5 a)·(0.5 b)
    float sig = 1.0f / (1.0f + expf(-s));
    out_sig[gw] = sig;
    float y   = labels[gw];
    float lp  = log1pf(expf(-fabsf(s)));
    float spm = fmaxf(-s, 0.0f) + lp;       // softplus(-s)
    float spp = fmaxf(s, 0.0f)  + lp;       // softplus(s)
    float w   = 1.0f + y;                   // POS_WEIGHT=2 -> 2y + (1-y)
    atomicAdd(bce_acc, w * (y * spm + (1.0f - y) * spp));
  }
}

// ---------------- SSL exp-sum GEMM (WMMA, streaming, async B tiles) ---------
// expsum[i] = sum_j exp( (N1[i]·N2[j]) / T ),  i,j in [0,4096)
__global__ __launch_bounds__(128)
void k_ssl(const float* __restrict__ N1, const float* __restrict__ N2,
           float* __restrict__ expsum) {
  __shared__ __align__(16) float ldsA[64 * LDSP];
  __shared__ __align__(16) float ldsB[2][16 * LDSP];
  const int tid  = threadIdx.x;
  const int wv   = tid >> 5;
  const int lane = tid & 31;
  const int half = lane >> 4;
  const int l16  = lane & 15;
  const int m0   = blockIdx.x * 64;
  const int c4   = (tid & 15) * 4;
  const int r8   = tid >> 4;

  for (int rr = r8; rr < 64; rr += 8)
    *(float4*)&ldsA[rr * LDSP + c4] =
        *(const float4*)(N1 + (size_t)(m0 + rr) * DD + c4);

  auto stageB = [&](int buf, int j0) {
#if HAS_ASYNC_LDS
    async_ld16(N2 + (size_t)(j0 + r8) * DD + c4,     &ldsB[buf][r8 * LDSP + c4]);
    async_ld16(N2 + (size_t)(j0 + r8 + 8) * DD + c4, &ldsB[buf][(r8 + 8) * LDSP + c4]);
#else
    *(float4*)&ldsB[buf][r8 * LDSP + c4] =
        *(const float4*)(N2 + (size_t)(j0 + r8) * DD + c4);
    *(float4*)&ldsB[buf][(r8 + 8) * LDSP + c4] =
        *(const float4*)(N2 + (size_t)(j0 + r8 + 8) * DD + c4);
#endif
  };

  v8f es = {};
  stageB(0, 0);
  for (int j0 = 0; j0 < BB; j0 += 16) {
    const int  cur  = (j0 >> 4) & 1;
    const bool more = (j0 + 16 < BB);
    if (more) {
      stageB(cur ^ 1, j0 + 16);
      wait_async<2>();
    } else {
      wait_async<0>();
    }
    __syncthreads();

    v8f s = {};
    const float* Ab = &ldsA[(size_t)(wv * 16 + l16) * LDSP];
    const float* Bb = &ldsB[cur][(size_t)l16 * LDSP];
#pragma unroll
    for (int k = 0; k < 64; k += 4) {
      const int kk = k + 2 * half;
      v2f a = *(const v2f*)(Ab + kk);
      v2f b = *(const v2f*)(Bb + kk);
      s = wmma4(a, b, s);
    }
#pragma unroll
    for (int r = 0; r < 8; ++r) es[r] += __expf(s[r] * INV_T);
    __syncthreads();
  }

#pragma unroll
  for (int r = 0; r < 8; ++r) {
    float v = es[r];
    v += __shfl_xor(v, 1);
    v += __shfl_xor(v, 2);
    v += __shfl_xor(v, 4);
    v += __shfl_xor(v, 8);   // reduce within 16-lane half (N dimension)
    if (l16 == 0) expsum[m0 + wv * 16 + 8 * half + r] = v;
  }
}

// ---------------- final scalar reduce ----------------
__global__ __launch_bounds__(256)
void k_final(const float* __restrict__ posr, const float* __restrict__ posd,
             const float* __restrict__ esr, const float* __restrict__ esd,
             const float* __restrict__ bce_acc, float* __restrict__ out_loss) {
  __shared__ float red[256];
  float a = 0.0f;
  for (int i = threadIdx.x; i < BB; i += 256)
    a += (logf(esr[i]) - posr[i]) + (logf(esd[i]) - posd[i]);
  red[threadIdx.x] = a;
  __syncthreads();
  for (int s = 128; s > 0; s >>= 1) {
    if (threadIdx.x < (unsigned)s) red[threadIdx.x] += red[threadIdx.x + s];
    __syncthreads();
  }
  if (threadIdx.x == 0) {
    float ssl = 0.05f * (red[0] / (float)BB);  // 0.05*(mean_r + mean_d)
    float bce = bce_acc[0] / (float)BB;
    out_loss[0] = bce + 0.3f * ssl;
  }
}

// ---------------- host launcher ----------------
extern "C" void kernel_launch(void* const* d_in, const int* in_sizes, int n_in,
                              void* d_out, int out_size, void* d_ws, size_t ws_size,
                              hipStream_t stream) {
  const float* Er0    = (const float*)d_in[0];
  const float* Ed0    = (const float*)d_in[1];
  const float* rec    = (const float*)d_in[2];
  const float* vals   = (const float*)d_in[3];
  const float* labels = (const float*)d_in[4];
  const int*   erow   = (const int*)d_in[5];
  const int*   ecol   = (const int*)d_in[6];
  const int*   drugs  = (const int*)d_in[7];
  const int*   dis    = (const int*)d_in[8];
  float* out = (float*)d_out;

  const size_t NE = (size_t)DRUG_NUM * DD;  // 512000
  const size_t NB = (size_t)BB * DD;        // 262144
  float* ws = (float*)d_ws;
  size_t o = 0;
  float* Zr0   = ws + o; o += NE;
  float* Zd0   = ws + o; o += NE;
  float* X     = ws + o; o += NE;
  float* Y     = ws + o; o += NE;
  float* SumEr = ws + o; o += NE;
  float* SumEd = ws + o; o += NE;
  float* SumGr = ws + o; o += NB;
  float* SumGd = ws + o; o += NB;
  float* N1r   = ws + o; o += NB;
  float* N2r   = ws + o; o += NB;
  float* N1d   = ws + o; o += NB;
  float* N2d   = ws + o; o += NB;
  float* posr  = ws + o; o += BB;
  float* posd  = ws + o; o += BB;
  float* esr   = ws + o; o += BB;
  float* esd   = ws + o; o += BB;
  float* scal  = ws + o; o += 8;            // scal[0] = bce accumulator

  k_zero<<<(int)((2 * NE + 255) / 256), 256, 0, stream>>>(Zr0, (int)(2 * NE));
  k_zero<<<1, 32, 0, stream>>>(scal, 8);

  // layer 0 sparse: Zr0 = A@Ed0 ; Zd0 = A^T@Er0
  k_spmm<<<(NNZ_ * 32 + 255) / 256, 256, 0, stream>>>(erow, ecol, vals, Ed0, Er0,
                                                      Zr0, Zd0, NNZ_);
  // X = Ed0+Zd0 (=SumEd init), Y = Er0+Zr0 (=SumEr init)
  k_addinit<<<(int)((NE + 255) / 256), 256, 0, stream>>>(Er0, Ed0, Zr0, Zd0, X, Y,
                                                         SumEr, SumEd, (int)NE);
  // layer 1 sparse: SumEr += A@Zd0 ; SumEd += A^T@Zr0
  k_spmm<<<(NNZ_ * 32 + 255) / 256, 256, 0, stream>>>(erow, ecol, vals, Zd0, Zr0,
                                                      SumEr, SumEd, NNZ_);
  // dense (single logical pass over rec): SumGr = Er0 + rec@X ; SumGd = Ed0 + rec^T@Y
  k_gemm<false><<<BB / 64, 128, 0, stream>>>(rec, X, Er0, SumGr);
  k_gemm<true ><<<BB / 64, 128, 0, stream>>>(rec, Y, Ed0, SumGd);

  // per-row fused: scores/sigmoid/BCE + normalization + pos terms
  k_rowfuse<<<BB * 32 / 256, 256, 0, stream>>>(SumEr, SumGr, SumEd, SumGd, labels,
                                               drugs, dis, N1r, N2r, N1d, N2d,
                                               posr, posd, scal, out + 1);
  // SSL denominators (streaming exp-sum GEMMs)
  k_ssl<<<BB / 64, 128, 0, stream>>>(N1r, N2r, esr);
  k_ssl<<<BB / 64, 128, 0, stream>>>(N1d, N2d, esd);

  // final loss scalar
  k_final<<<1, 256, 0, stream>>>(posr, posd, esr, esd, scal, out);
}